// GraphAttnLayer_70196945486348
// MI455X (gfx1250) — compile-verified
//
#include <hip/hip_runtime.h>
#include <math.h>

typedef __attribute__((ext_vector_type(2))) float v2f;
typedef __attribute__((ext_vector_type(8))) float v8f;

#define IN_F 128
#define OUT_F 128
#define NH 4
#define HD 32
#define NEG_SLOPE 0.2f
#define LN_EPS 1e-12f
#define KSTEPS (IN_F / 4)   // 32 WMMA k-steps

// ---------------------------------------------------------------------------
// Kernel 1: fused dual projection  fs = x@W_src + b_src, fd = x@W_dst + b_dst
// fp32 WMMA 16x16x4. Block = 512 threads (16 waves); wave w -> column tile w
// (tiles 0..7 -> fs, 8..15 -> fd). Persistent blocks: B fragments (64 floats
// per lane, covering all K) are loaded into VGPRs once, then the block loops
// over M-tiles with a pure  A-load(b64) -> v_wmma  inner loop.
// ---------------------------------------------------------------------------
__global__ void proj_wmma_kernel(const float* __restrict__ x,
                                 const float* __restrict__ Wsrc,
                                 const float* __restrict__ bsrc,
                                 const float* __restrict__ Wdst,
                                 const float* __restrict__ bdst,
                                 float* __restrict__ fs,
                                 float* __restrict__ fd,
                                 int n, int ntiles) {
    const int wave = threadIdx.x >> 5;   // 0..15
    const int lane = threadIdx.x & 31;
    const int half = lane >> 4;          // 0: K={0,1}, 1: K={2,3}
    const int l16  = lane & 15;

    const bool is_dst = (wave >= 8);
    const int  ntile  = is_dst ? (wave - 8) : wave;
    const float* __restrict__ W    = is_dst ? Wdst : Wsrc;
    const float* __restrict__ bias = is_dst ? bdst : bsrc;
    float* __restrict__ out        = is_dst ? fd : fs;

    const int col = ntile * 16 + l16;    // B/C column this lane feeds
    const float bc = bias[col];

    // Preload all B fragments for this column tile (whole K) into registers.
    v2f bfrag[KSTEPS];
    #pragma unroll
    for (int kk = 0; kk < KSTEPS; ++kk) {
        const int k = kk * 4 + half * 2;
        bfrag[kk].x = W[(size_t)k * OUT_F + col];
        bfrag[kk].y = W[(size_t)(k + 1) * OUT_F + col];
    }

    for (int mtile = blockIdx.x; mtile < ntiles; mtile += gridDim.x) {
        int row = mtile * 16 + l16;      // A-matrix row this lane feeds
        if (row >= n) row = n - 1;       // clamp (keeps EXEC all-ones for WMMA)
        const float* __restrict__ xr = x + (size_t)row * IN_F;

        v8f acc = {};
        #pragma unroll
        for (int kk = 0; kk < KSTEPS; ++kk) {
            const int k = kk * 4 + half * 2;
            v2f a = { xr[k], xr[k + 1] };
            acc = __builtin_amdgcn_wmma_f32_16x16x4_f32(
                /*neg_a=*/false, a, /*neg_b=*/false, bfrag[kk],
                /*c_mod=*/(short)0, acc, /*reuse_a=*/false, /*reuse_b=*/false);
        }

        // C layout: VGPR i -> row i (lanes 0-15) / row 8+i (lanes 16-31)
        const int rbase = mtile * 16 + half * 8;
        if ((mtile + 1) * 16 <= n) {     // uniform fast path: no per-lane guard
            #pragma unroll
            for (int i = 0; i < 8; ++i)
                out[(size_t)(rbase + i) * OUT_F + col] = acc[i] + bc;
        } else {
            #pragma unroll
            for (int i = 0; i < 8; ++i) {
                const int r = rbase + i;
                if (r < n) out[(size_t)r * OUT_F + col] = acc[i] + bc;
            }
        }
    }
}

// ---------------------------------------------------------------------------
// Kernel 2: init accumulators. out = 0, sumv = 0, maxi = ordered-int -inf.
// ---------------------------------------------------------------------------
__global__ void init_kernel(float* __restrict__ out,
                            float* __restrict__ sumv,
                            int*   __restrict__ maxi,
                            int n) {
    const int idx = blockIdx.x * blockDim.x + threadIdx.x;
    if (idx < n * OUT_F) out[idx] = 0.0f;
    if (idx < n * NH) { sumv[idx] = 0.0f; maxi[idx] = (int)0x80000000; }
}

__device__ __forceinline__ float lrelu(float v) {
    return v > 0.0f ? v : NEG_SLOPE * v;
}

// ordered-int mapping: monotone float -> signed int
__device__ __forceinline__ int f2ord(float f) {
    int m = __float_as_int(f);
    return m >= 0 ? m : (m ^ 0x7fffffff);
}
__device__ __forceinline__ float ord2f(int m) {
    return __int_as_float(m >= 0 ? m : (m ^ 0x7fffffff));
}

// ---------------------------------------------------------------------------
// Kernel 3: per-edge logits + segment max. One wave per edge.
// Lane i holds float4 chunk i of the 128-float rows; head h = lane>>3.
// ---------------------------------------------------------------------------
__global__ void edge_logits_kernel(const float4* __restrict__ fs4,
                                   const float4* __restrict__ fd4,
                                   const float4* __restrict__ attn4,
                                   const int* __restrict__ src,
                                   const int* __restrict__ dst,
                                   float* __restrict__ logits,
                                   int*   __restrict__ maxi,
                                   int E) {
    const int wid  = blockIdx.x * (blockDim.x >> 5) + (threadIdx.x >> 5);
    const int lane = threadIdx.x & 31;
    if (wid >= E) return;

    const int s = src[wid];
    const int t = dst[wid];
    const float4 a = fs4[(size_t)s * 32 + lane];
    const float4 b = fd4[(size_t)t * 32 + lane];
    const float4 w = attn4[lane];

    float p = w.x * lrelu(a.x + b.x)
            + w.y * lrelu(a.y + b.y)
            + w.z * lrelu(a.z + b.z)
            + w.w * lrelu(a.w + b.w);

    // reduce within each 8-lane head group
    p += __shfl_xor(p, 1, 32);
    p += __shfl_xor(p, 2, 32);
    p += __shfl_xor(p, 4, 32);

    if ((lane & 7) == 0) {
        const int h = lane >> 3;
        logits[(size_t)wid * NH + h] = p;
        atomicMax(&maxi[t * NH + h], f2ord(p));
    }
}

// ---------------------------------------------------------------------------
// Kernel 4: ex = exp(logit - max[dst]); sumv[dst] += ex. Thread per (e,h).
// ---------------------------------------------------------------------------
__global__ void edge_exp_kernel(float* __restrict__ exl,
                                const int* __restrict__ dst,
                                const int* __restrict__ maxi,
                                float* __restrict__ sumv,
                                int E4) {
    const int idx = blockIdx.x * blockDim.x + threadIdx.x;
    if (idx >= E4) return;
    const int e = idx >> 2;
    const int h = idx & 3;
    const int t = dst[e];
    const float M = ord2f(maxi[t * NH + h]);
    const float ex = expf(exl[idx] - M);
    exl[idx] = ex;
    atomicAdd(&sumv[t * NH + h], ex);
}

// ---------------------------------------------------------------------------
// Kernel 5: rst[dst] += alpha * fs[src]. One wave per edge.
// ---------------------------------------------------------------------------
__global__ void edge_aggr_kernel(const float4* __restrict__ fs4,
                                 const float* __restrict__ exl,
                                 const float* __restrict__ sumv,
                                 const int* __restrict__ src,
                                 const int* __restrict__ dst,
                                 float* __restrict__ out,
                                 int E) {
    const int wid  = blockIdx.x * (blockDim.x >> 5) + (threadIdx.x >> 5);
    const int lane = threadIdx.x & 31;
    if (wid >= E) return;

    const int s = src[wid];
    const int t = dst[wid];
    const int h = lane >> 3;
    const float alpha = exl[(size_t)wid * NH + h] / sumv[t * NH + h];
    const float4 a = fs4[(size_t)s * 32 + lane];

    float* o = out + (size_t)t * OUT_F + lane * 4;
    atomicAdd(o + 0, alpha * a.x);
    atomicAdd(o + 1, alpha * a.y);
    atomicAdd(o + 2, alpha * a.z);
    atomicAdd(o + 3, alpha * a.w);
}

// ---------------------------------------------------------------------------
// Kernel 6: +out_bias, LayerNorm (biased var, eps inside sqrt), ELU. In place.
// One wave per node row of 128.
// ---------------------------------------------------------------------------
__global__ void finalize_kernel(float* __restrict__ out,
                                const float4* __restrict__ bias4,
                                const float4* __restrict__ lw4,
                                const float4* __restrict__ lb4,
                                int n) {
    const int wid  = blockIdx.x * (blockDim.x >> 5) + (threadIdx.x >> 5);
    const int lane = threadIdx.x & 31;
    if (wid >= n) return;

    float4* o4 = (float4*)out + (size_t)wid * 32;
    float4 v = o4[lane];
    const float4 b = bias4[lane];
    v.x += b.x; v.y += b.y; v.z += b.z; v.w += b.w;

    float sum = v.x + v.y + v.z + v.w;
    #pragma unroll
    for (int m = 1; m < 32; m <<= 1) sum += __shfl_xor(sum, m, 32);
    const float mean = sum * (1.0f / 128.0f);

    const float dx = v.x - mean, dy = v.y - mean, dz = v.z - mean, dw = v.w - mean;
    float ss = dx * dx + dy * dy + dz * dz + dw * dw;
    #pragma unroll
    for (int m = 1; m < 32; m <<= 1) ss += __shfl_xor(ss, m, 32);
    const float inv = rsqrtf(ss * (1.0f / 128.0f) + LN_EPS);

    const float4 lw = lw4[lane];
    const float4 lb = lb4[lane];
    float4 y;
    y.x = lw.x * dx * inv + lb.x;
    y.y = lw.y * dy * inv + lb.y;
    y.z = lw.z * dz * inv + lb.z;
    y.w = lw.w * dw * inv + lb.w;
    // ELU (alpha=1)
    y.x = y.x > 0.0f ? y.x : expm1f(y.x);
    y.y = y.y > 0.0f ? y.y : expm1f(y.y);
    y.z = y.z > 0.0f ? y.z : expm1f(y.z);
    y.w = y.w > 0.0f ? y.w : expm1f(y.w);

    o4[lane] = y;
}

// ---------------------------------------------------------------------------
extern "C" void kernel_launch(void* const* d_in, const int* in_sizes, int n_in,
                              void* d_out, int out_size, void* d_ws, size_t ws_size,
                              hipStream_t stream) {
    const float* x     = (const float*)d_in[0];
    const float* Wsrc  = (const float*)d_in[1];
    const float* bsrc  = (const float*)d_in[2];
    const float* Wdst  = (const float*)d_in[3];
    const float* bdst  = (const float*)d_in[4];
    const float* attn  = (const float*)d_in[5];
    const float* obias = (const float*)d_in[6];
    const float* lnw   = (const float*)d_in[7];
    const float* lnb   = (const float*)d_in[8];
    const int*   src   = (const int*)d_in[9];
    const int*   dst   = (const int*)d_in[10];

    const int n = in_sizes[0] / IN_F;   // 50000
    const int E = in_sizes[9];          // 800000
    float* out = (float*)d_out;

    // workspace layout (floats): fs | fd | ex/logits | maxi | sumv
    float* fs   = (float*)d_ws;
    float* fd   = fs + (size_t)n * OUT_F;
    float* exl  = fd + (size_t)n * OUT_F;
    int*   maxi = (int*)(exl + (size_t)E * NH);
    float* sumv = (float*)(maxi + (size_t)n * NH);

    const int ntiles = (n + 15) / 16;                 // 3125
    int nblocks = (ntiles + 7) / 8;                   // ~8 M-tiles per block
    if (nblocks < 1) nblocks = 1;

    proj_wmma_kernel<<<nblocks, 512, 0, stream>>>(x, Wsrc, bsrc, Wdst, bdst,
                                                  fs, fd, n, ntiles);
    init_kernel<<<(n * OUT_F + 255) / 256, 256, 0, stream>>>(out, sumv, maxi, n);
    edge_logits_kernel<<<(E + 7) / 8, 256, 0, stream>>>(
        (const float4*)fs, (const float4*)fd, (const float4*)attn, src, dst, exl, maxi, E);
    edge_exp_kernel<<<(E * NH + 255) / 256, 256, 0, stream>>>(exl, dst, maxi, sumv, E * NH);
    edge_aggr_kernel<<<(E + 7) / 8, 256, 0, stream>>>(
        (const float4*)fs, exl, sumv, src, dst, out, E);
    finalize_kernel<<<(n + 7) / 8, 256, 0, stream>>>(
        out, (const float4*)obias, (const float4*)lnw, (const float4*)lnb, n);
}